// FGPM_66151086293031
// MI455X (gfx1250) — compile-verified
//
#include <hip/hip_runtime.h>
#include <math.h>
#include <stdint.h>

// Geometry kernel: 6 x (B,2) f32 points -> (B,55) f32 [15 dists | 20 angles | 20 areas]
// Strategy: one thread per row computes all 55 features into an LDS image of the
// block's contiguous output region; the block then streams LDS -> HBM with
// CDNA5 async stores (global_store_async_from_lds_b128, ASYNCcnt), giving fully
// coalesced 512B-per-wave-instruction writes with no VGPR round trip.

#define ROWS 256      // rows per block == threads per block (8 waves on CDNA5)
#define NFEAT 55

__global__ __launch_bounds__(ROWS) void FGPM_geom_kernel(
    const float* __restrict__ q0, const float* __restrict__ q1,
    const float* __restrict__ q2, const float* __restrict__ q3,
    const float* __restrict__ q4, const float* __restrict__ q5,
    float* __restrict__ out, int B)
{
    __shared__ __align__(16) float smem[ROWS * NFEAT];   // 56320 B

    const int row0 = blockIdx.x * ROWS;
    const int row  = row0 + threadIdx.x;

    if (row < B) {
        const float* ptrs[6] = {q0, q1, q2, q3, q4, q5};
        float px[6], py[6];
        #pragma unroll
        for (int k = 0; k < 6; ++k) {
            float2 v = reinterpret_cast<const float2*>(ptrs[k])[row]; // global_load_b64, coalesced
            px[k] = v.x; py[k] = v.y;
        }

        float* o = &smem[threadIdx.x * NFEAT];  // stride 55 words: bank-conflict-free (gcd(55,64)=1)

        // ---- 15 pairwise distances (lexicographic i<j) ----
        constexpr int PI[15] = {0,0,0,0,0,1,1,1,1,2,2,2,3,3,4};
        constexpr int PJ[15] = {1,2,3,4,5,2,3,4,5,3,4,5,4,5,5};
        #pragma unroll
        for (int p = 0; p < 15; ++p) {
            float dx = px[PI[p]] - px[PJ[p]];
            float dy = py[PI[p]] - py[PJ[p]];
            o[p] = sqrtf(dx * dx + dy * dy);
        }

        // ---- 20 triplets (lexicographic i<j<k): angle at vertex j, triangle area ----
        constexpr int TI[20] = {0,0,0,0,0,0,0,0,0,0,1,1,1,1,1,1,2,2,2,3};
        constexpr int TJ[20] = {1,1,1,1,2,2,2,3,3,4,2,2,2,3,3,4,3,3,4,4};
        constexpr int TK[20] = {2,3,4,5,3,4,5,4,5,5,3,4,5,4,5,5,4,5,5,5};
        #pragma unroll
        for (int t = 0; t < 20; ++t) {
            const int i = TI[t], j = TJ[t], k = TK[t];
            float v1x = px[i] - px[j], v1y = py[i] - py[j];
            float v2x = px[k] - px[j], v2y = py[k] - py[j];
            float dot = v1x * v2x + v1y * v2y;
            float n2  = (v1x * v1x + v1y * v1y) * (v2x * v2x + v2y * v2y);
            float c   = dot * rsqrtf(n2);                 // v_rsq_f32: avoids sqrt+sqrt+div
            c = fminf(1.0f, fmaxf(-1.0f, c));
            o[15 + t] = acosf(c) * 57.29577951308232f;    // degrees
            o[35 + t] = 0.5f * fabsf(px[i] * (py[j] - py[k]) +
                                     px[j] * (py[k] - py[i]) +
                                     px[k] * (py[i] - py[j]));
        }
    }

    __syncthreads();  // all LDS writes visible before async engine reads them

    int nrows = B - row0; if (nrows > ROWS) nrows = ROWS;
    if (nrows <= 0) return;
    const int nfl = nrows * NFEAT;          // floats staged in this block
    const int n4  = nfl >> 2;               // 16-byte chunks

    float* gbase = out + (size_t)row0 * NFEAT;  // 16B-aligned: row0*55*4 is a multiple of 64*55*4

    // CDNA5 async LDS->global stream: 16B per lane, no VGPR data movement, tracked on ASYNCcnt.
    for (int idx = threadIdx.x; idx < n4; idx += ROWS) {
        uint64_t gaddr = (uint64_t)(uintptr_t)(gbase + (idx << 2));
        uint32_t laddr = (uint32_t)(uintptr_t)(&smem[idx << 2]);  // low 32 bits = LDS byte offset
        asm volatile("global_store_async_from_lds_b128 %0, %1, off"
                     :: "v"(gaddr), "v"(laddr)
                     : "memory");
    }
    // generic scalar tail (empty for B = 2,000,000: 128*55 is a multiple of 4)
    for (int idx = (n4 << 2) + threadIdx.x; idx < nfl; idx += ROWS) {
        gbase[idx] = smem[idx];
    }

    asm volatile("s_wait_asynccnt 0x0" ::: "memory");  // drain ASYNCcnt before wave exit
}

extern "C" void kernel_launch(void* const* d_in, const int* in_sizes, int n_in,
                              void* d_out, int out_size, void* d_ws, size_t ws_size,
                              hipStream_t stream) {
    (void)n_in; (void)out_size; (void)d_ws; (void)ws_size;
    const int B = in_sizes[0] / 2;                 // inputs are (B,2) f32
    const int grid = (B + ROWS - 1) / ROWS;        // 7813 for B = 2,000,000
    FGPM_geom_kernel<<<grid, ROWS, 0, stream>>>(
        (const float*)d_in[0], (const float*)d_in[1], (const float*)d_in[2],
        (const float*)d_in[3], (const float*)d_in[4], (const float*)d_in[5],
        (float*)d_out, B);
}